// SAMPNConv_7876970021288
// MI455X (gfx1250) — compile-verified
//
#include <hip/hip_runtime.h>

typedef __attribute__((ext_vector_type(2))) float v2f;
typedef __attribute__((ext_vector_type(8))) float v8f;

#define FEATS    128
#define EDGE_DIM 16

// LDS layout: reorganized weights, row-pair major.
//   element (row r, col n) of W  ->  pair p = r>>1, parity i = r&1
//   dword offset = REGION + p*256 + ( ((n<<1 | i) + ((p&1)<<5)) & 255 )
// The ((p&1)<<5) rotate makes even-p (lanes 0-15) and odd-p (lanes 16-31)
// halves of one B-fragment load hit opposite 32-bank halves -> conflict-free.
#define WIN_PAIRS   72            // 144 rows of W_in
#define WHID_PAIRS  64            // 128 rows of W_hid
#define HID_BASE_DW (WIN_PAIRS * 256)                  // 18432 dwords
#define LDS_DWORDS  (HID_BASE_DW + WHID_PAIRS * 256)   // 34816 dwords = 139264 B

// ---------------------------------------------------------------------------
// Kernel 1: zero the node_sum accumulator (lives in d_ws, poisoned by harness)
// ---------------------------------------------------------------------------
__global__ void sampn_zero_kernel(float* __restrict__ p, int n) {
    int i = blockIdx.x * blockDim.x + threadIdx.x;
    int stride = gridDim.x * blockDim.x;
    for (; i < n; i += stride) p[i] = 0.0f;
}

// ---------------------------------------------------------------------------
// Kernel 2: segment_sum(messages, dst) -> node_sum  (scatter-add, f32 atomics)
// 25.6MB accumulator stays resident in the 192MB L2, so atomic RMW is cheap.
// ---------------------------------------------------------------------------
__global__ void sampn_scatter_kernel(const float* __restrict__ messages,
                                     const int*   __restrict__ dst,
                                     float*       __restrict__ node_sum,
                                     int n_edges) {
    int idx = blockIdx.x * blockDim.x + threadIdx.x;
    int e = idx >> 5;          // edge index
    int c = (idx & 31) << 2;   // feature chunk start (0..124 step 4)
    if (e >= n_edges) return;
    const float4 m = *(const float4*)(messages + (size_t)e * FEATS + c);
    float* base = node_sum + (size_t)dst[e] * FEATS + c;
    atomicAdd(base + 0, m.x);
    atomicAdd(base + 1, m.y);
    atomicAdd(base + 2, m.z);
    atomicAdd(base + 3, m.w);
}

// ---------------------------------------------------------------------------
// Kernel 3: per-edge GEMMs via V_WMMA_F32_16X16X4_F32, weights in LDS.
//   mess = [nf[src], ef] @ W_in + (node_sum[src] - msg) @ W_hid + b_in + b_hid
//
// f32 WMMA operand layouts (ISA §7.12.2), m=lane&15, h=lane>>4, kk=4k+2h:
//   A 16x4 : v.x = A[m][kk],    v.y = A[m][kk+1]
//   B 4x16 : v.x = B[kk][n],    v.y = B[kk+1][n]   (64-bit LDS load)
//   C/D    : v8f, component r -> row (r + 8*h), col n
// Weights staged to LDS once per block with global_load_async_to_lds_b32
// (ASYNCcnt path). All 8 B fragments of a k-step are loaded into b[8] BEFORE
// the 8-WMMA chain so the scheduler can burst-issue the ds loads and stagger
// the dscnt waits instead of serializing load->wait0->wmma.
// EXEC is all-1s for every WMMA (tile guards are wave-uniform).
// ---------------------------------------------------------------------------
__global__ __launch_bounds__(256) void sampn_gemm_kernel(
    const float* __restrict__ node_feats,
    const float* __restrict__ edge_feats,
    const float* __restrict__ messages,
    const float* __restrict__ W_in,    // [144][128] row-major
    const float* __restrict__ b_in,    // [128]
    const float* __restrict__ W_hid,   // [128][128] row-major
    const float* __restrict__ b_hid,   // [128]
    const int*   __restrict__ src,
    const float* __restrict__ node_sum,
    float*       __restrict__ out,     // [E][128]
    int ntiles) {

    extern __shared__ float lds[];     // LDS_DWORDS floats, offset 0 in LDS

    const int tid  = threadIdx.x;
    const int lane = tid & 31;
    const int wave = tid >> 5;
    const int m    = lane & 15;        // row within 16-edge tile / col within N-tile
    const int h    = lane >> 4;        // half: 0 -> lanes 0-15, 1 -> lanes 16-31

    // ---- stage reorganized W_in / W_hid into LDS via async-to-LDS path ----
    for (int idx = tid; idx < 144 * FEATS; idx += 256) {
        const int r = idx >> 7, n = idx & 127;
        const int p = r >> 1, i = r & 1;
        const unsigned dw = (unsigned)(p * 256 + ((((n << 1) | i) + ((p & 1) << 5)) & 255));
        const unsigned lds_byte = dw << 2;
        const float* g = W_in + idx;
        asm volatile("global_load_async_to_lds_b32 %0, %1, off"
                     :: "v"(lds_byte), "v"(g) : "memory");
    }
    for (int idx = tid; idx < FEATS * FEATS; idx += 256) {
        const int r = idx >> 7, n = idx & 127;
        const int p = r >> 1, i = r & 1;
        const unsigned dw = (unsigned)(HID_BASE_DW + p * 256 +
                                       ((((n << 1) | i) + ((p & 1) << 5)) & 255));
        const unsigned lds_byte = dw << 2;
        const float* g = W_hid + idx;
        asm volatile("global_load_async_to_lds_b32 %0, %1, off"
                     :: "v"(lds_byte), "v"(g) : "memory");
    }
    asm volatile("s_wait_asynccnt 0" ::: "memory");
    __syncthreads();

    // per-lane B column dword rotation, constant across k (p&1 == h everywhere)
    int cdw[8];
    float bias[8];
    #pragma unroll
    for (int t = 0; t < 8; ++t) {
        const int n = t * 16 + m;
        cdw[t]  = (((n << 1) + (h << 5)) & 255);
        bias[t] = b_in[n] + b_hid[n];
    }

    const v8f vzero = {0.f, 0.f, 0.f, 0.f, 0.f, 0.f, 0.f, 0.f};

    // ---- each wave processes 8 tiles (block covers 64 tiles) ----
    for (int tp = 0; tp < 8; ++tp) {
        const int tile = blockIdx.x * 64 + wave * 8 + tp;
        if (tile >= ntiles) break;                 // wave-uniform

        const int e = tile * 16 + m;               // this lane's edge
        const int s = src[e];                      // gather index

        const float* nf = node_feats + (size_t)s * FEATS;
        const float* ef = edge_feats + (size_t)e * EDGE_DIM;
        const float* ns = node_sum   + (size_t)s * FEATS;
        const float* mg = messages   + (size_t)e * FEATS;

        v8f acc[8];
        #pragma unroll
        for (int t = 0; t < 8; ++t) acc[t] = vzero;

        // input path, node_feats portion of W_in: K = 0..127 (pairs p = 2k+h)
        for (int k = 0; k < 32; ++k) {
            const int kk = k * 4 + h * 2;
            const v2f a = *(const v2f*)(nf + kk);
            const float* lp = &lds[(2 * k + h) * 256];
            v2f b[8];
            #pragma unroll
            for (int t = 0; t < 8; ++t) b[t] = *(const v2f*)(lp + cdw[t]);
            #pragma unroll
            for (int t = 0; t < 8; ++t)
                acc[t] = __builtin_amdgcn_wmma_f32_16x16x4_f32(
                    false, a, false, b[t], (short)0, acc[t], false, false);
        }
        // input path, edge_feats portion of W_in: K = 128..143 (pairs p = 64+2k+h)
        for (int k = 0; k < 4; ++k) {
            const int kk = k * 4 + h * 2;
            const v2f a = *(const v2f*)(ef + kk);
            const float* lp = &lds[(64 + 2 * k + h) * 256];
            v2f b[8];
            #pragma unroll
            for (int t = 0; t < 8; ++t) b[t] = *(const v2f*)(lp + cdw[t]);
            #pragma unroll
            for (int t = 0; t < 8; ++t)
                acc[t] = __builtin_amdgcn_wmma_f32_16x16x4_f32(
                    false, a, false, b[t], (short)0, acc[t], false, false);
        }
        // hidden path: A = node_sum[src] - messages, W_hid: K = 0..127
        for (int k = 0; k < 32; ++k) {
            const int kk = k * 4 + h * 2;
            const v2f av = *(const v2f*)(ns + kk);
            const v2f mv = *(const v2f*)(mg + kk);
            v2f a; a.x = av.x - mv.x; a.y = av.y - mv.y;
            const float* lp = &lds[HID_BASE_DW + (2 * k + h) * 256];
            v2f b[8];
            #pragma unroll
            for (int t = 0; t < 8; ++t) b[t] = *(const v2f*)(lp + cdw[t]);
            #pragma unroll
            for (int t = 0; t < 8; ++t)
                acc[t] = __builtin_amdgcn_wmma_f32_16x16x4_f32(
                    false, a, false, b[t], (short)0, acc[t], false, false);
        }

        // epilogue: bias + store (lanes 0-15 -> rows 0-7, lanes 16-31 -> rows 8-15)
        #pragma unroll
        for (int t = 0; t < 8; ++t) {
            const int n = t * 16 + m;
            #pragma unroll
            for (int r = 0; r < 8; ++r) {
                const int row = tile * 16 + h * 8 + r;
                out[(size_t)row * FEATS + n] = acc[t][r] + bias[t];
            }
        }
    }
}

// ---------------------------------------------------------------------------
extern "C" void kernel_launch(void* const* d_in, const int* in_sizes, int n_in,
                              void* d_out, int out_size, void* d_ws, size_t ws_size,
                              hipStream_t stream) {
    const float* node_feats = (const float*)d_in[0];
    const float* edge_feats = (const float*)d_in[1];
    const float* messages   = (const float*)d_in[2];
    const float* W_in       = (const float*)d_in[3];
    const float* b_in       = (const float*)d_in[4];
    const float* W_hid      = (const float*)d_in[5];
    const float* b_hid      = (const float*)d_in[6];
    const int*   src        = (const int*)d_in[7];
    const int*   dst        = (const int*)d_in[8];
    float* out      = (float*)d_out;
    float* node_sum = (float*)d_ws;   // n_nodes * FEATS floats (25.6 MB)

    const int n_nodes = in_sizes[0] / FEATS;
    const int n_edges = in_sizes[2] / FEATS;
    const int nsum    = n_nodes * FEATS;

    // 1) zero accumulator (grid-stride)
    {
        int blocks = (nsum + 256 * 4 - 1) / (256 * 4);
        sampn_zero_kernel<<<blocks, 256, 0, stream>>>(node_sum, nsum);
    }
    // 2) scatter-add messages -> node_sum
    {
        long long threads = (long long)n_edges * 32;
        int blocks = (int)((threads + 255) / 256);
        sampn_scatter_kernel<<<blocks, 256, 0, stream>>>(messages, dst, node_sum, n_edges);
    }
    // 3) WMMA GEMM kernel: weights async-staged to LDS, 64 tiles per block
    {
        int ntiles = n_edges / 16;
        int blocks = (ntiles + 63) / 64;
        size_t lds_bytes = (size_t)LDS_DWORDS * sizeof(float);   // 139264 B
        sampn_gemm_kernel<<<blocks, 256, lds_bytes, stream>>>(
            node_feats, edge_feats, messages, W_in, b_in, W_hid, b_hid,
            src, node_sum, out, ntiles);
    }
}